// LSSM_85186381349363
// MI455X (gfx1250) — compile-verified
//
#include <hip/hip_runtime.h>
#include <hip/hip_bf16.h>

// Chunked parallel linear SSM scan for MI455X (gfx1250), fp32 WMMA path.
//   h_{cT+t} = A^t h0_c + sum_{j<t} A^{t-1-j} b u[cT+j],   T = 64, 256 chunks.
// Workspace layout (floats):
//   Vmat : 64*64          (Vmat[k][n] = (A^k b)[n])
//   PT   : 64*4096        (PT[m][t*64+n] = (A^t)[n][m])
//   A64  : 64*64          ((A^64)[n][m])
//   S    : 256*32*64      (per-chunk input accumulation)
//   H0   : 256*32*64      (per-chunk initial states)
// ~5.3 MB total.

typedef __attribute__((ext_vector_type(2))) float v2f;
typedef __attribute__((ext_vector_type(8))) float v8f;

#define NS   64      // state dim
#define TT   64      // chunk length
#define NB   32      // batch
#define NCH  256     // chunks = L / TT
#define LSEQ 16384
#define PAD  68      // LDS row stride (dwords): 68 mod 64 = 4 -> conflict-free lane strides

__device__ __forceinline__ v8f wmma4(v2f a, v2f b, v8f c) {
  // V_WMMA_F32_16X16X4_F32 : D(16x16,f32) = A(16x4,f32) x B(4x16,f32) + C
  return __builtin_amdgcn_wmma_f32_16x16x4_f32(
      /*neg_a=*/false, a, /*neg_b=*/false, b,
      /*c_mod=*/(short)0, c, /*reuse_a=*/false, /*reuse_b=*/false);
}

// ---------------------------------------------------------------- setup -----
// Single block: iterate cur = A^k, emitting Vmat[k], PT[:, k*64:(k+1)*64], A64.
__global__ void lssm_setup(const float* __restrict__ A, const float* __restrict__ bvec,
                           float* __restrict__ Vmat, float* __restrict__ PT,
                           float* __restrict__ A64) {
  __shared__ float sA[NS * NS];
  __shared__ float cur[NS * NS];
  __shared__ float nxt[NS * NS];
  const int tid = threadIdx.x;
  for (int i = tid; i < NS * NS; i += 256) sA[i] = A[i];
  for (int i = tid; i < NS * NS; i += 256) cur[i] = ((i >> 6) == (i & 63)) ? 1.f : 0.f;
  __syncthreads();
  for (int k = 0; k < TT; ++k) {
    // emit PT[m][k*64+n] = cur[n][m]
    for (int i = tid; i < NS * NS; i += 256) {
      int n = i >> 6, m = i & 63;
      PT[m * (TT * NS) + k * NS + n] = cur[i];
    }
    // Vmat[k][n] = cur[n][:] . b
    if (tid < NS) {
      float s = 0.f;
      for (int m = 0; m < NS; ++m) s += cur[tid * NS + m] * bvec[m];
      Vmat[k * NS + tid] = s;
    }
    // nxt = A @ cur
    for (int i = tid; i < NS * NS; i += 256) {
      int n = i >> 6, m = i & 63;
      float s = 0.f;
      for (int kk = 0; kk < NS; ++kk) s += sA[n * NS + kk] * cur[kk * NS + m];
      nxt[i] = s;
    }
    __syncthreads();
    for (int i = tid; i < NS * NS; i += 256) cur[i] = nxt[i];
    __syncthreads();
  }
  for (int i = tid; i < NS * NS; i += 256) A64[i] = cur[i];  // cur = A^64
}

// ----------------------------------------------------- per-chunk input sums --
// S_c(32x64) = U_c(32x64) @ Vrev(64x64),  Vrev[j] = Vmat[63-j].  WMMA, 8 tiles.
__global__ void lssm_chunksums(const float* __restrict__ u, const float* __restrict__ Vmat,
                               float* __restrict__ S) {
  const int c = blockIdx.x;
  __shared__ float Uc[NB][PAD];
  __shared__ float Vr[TT][PAD];
  const int tid = threadIdx.x;
  for (int i = tid; i < NB * TT; i += 256) {
    int b = i >> 6, j = i & 63;
    Uc[b][j] = u[(size_t)b * LSEQ + c * TT + j];
  }
  for (int i = tid; i < TT * NS; i += 256) {
    int j = i >> 6, n = i & 63;
    Vr[j][n] = Vmat[(TT - 1 - j) * NS + n];
  }
  __syncthreads();
  const int wave = tid >> 5, lane = tid & 31;
  const int l15 = lane & 15, khalf = (lane >> 4) << 1;
  const int mTile = wave >> 2, nTile = wave & 3;  // 2 x 4 output tiles
  v8f acc = {};
  for (int k0 = 0; k0 < TT; k0 += 4) {
    const int kb = k0 + khalf;
    v2f a, bb;
    const int br = mTile * 16 + l15;
    a.x = Uc[br][kb];      a.y = Uc[br][kb + 1];
    const int col = nTile * 16 + l15;
    bb.x = Vr[kb][col];    bb.y = Vr[kb + 1][col];
    acc = wmma4(a, bb, acc);
  }
  const int bofs = (lane >= 16) ? 8 : 0;
  for (int r = 0; r < 8; ++r) {
    const int b = mTile * 16 + r + bofs;
    const int n = nTile * 16 + l15;
    S[(size_t)c * (NB * NS) + b * NS + n] = acc[r];
  }
}

// --------------------------------------------------------- sequential carry --
// h0_{c+1} = A^64 h0_c + S_c.  255 short steps, single block.
__global__ void lssm_carry(const float* __restrict__ A64g, const float* __restrict__ S,
                           float* __restrict__ H0) {
  __shared__ float sA[NS * NS];
  __shared__ float h[NB * NS];
  __shared__ float hn[NB * NS];
  const int tid = threadIdx.x;
  for (int i = tid; i < NS * NS; i += 256) sA[i] = A64g[i];
  for (int i = tid; i < NB * NS; i += 256) { h[i] = 0.f; H0[i] = 0.f; }
  __syncthreads();
  for (int c = 0; c < NCH - 1; ++c) {
    for (int i = tid; i < NB * NS; i += 256) {
      int b = i >> 6, n = i & 63;
      float s = S[(size_t)c * (NB * NS) + i];
      for (int m = 0; m < NS; ++m) s += sA[n * NS + m] * h[b * NS + m];
      hn[i] = s;
      H0[(size_t)(c + 1) * (NB * NS) + i] = s;
    }
    __syncthreads();
    for (int i = tid; i < NB * NS; i += 256) h[i] = hn[i];
    __syncthreads();
  }
}

// ----------------------------------------------------- local Toeplitz matmul --
// H_local(64x64) = Um(64x64,lower-tri Toeplitz of u) @ Vmat(64x64).  WMMA.
__global__ void lssm_local(const float* __restrict__ u, const float* __restrict__ Vmat,
                           float* __restrict__ states) {
  const int c = blockIdx.x, b = blockIdx.y;
  __shared__ float Um[TT][PAD];
  __shared__ float Vm[TT][PAD];
  __shared__ float uloc[TT];
  const int tid = threadIdx.x;
  if (tid < TT) uloc[tid] = u[(size_t)b * LSEQ + c * TT + tid];
  for (int i = tid; i < TT * NS; i += 256) Vm[i >> 6][i & 63] = Vmat[i];
  __syncthreads();
  for (int i = tid; i < TT * TT; i += 256) {
    int t = i >> 6, k = i & 63;
    Um[t][k] = (k < t) ? uloc[t - 1 - k] : 0.f;
  }
  __syncthreads();
  const int wave = tid >> 5, lane = tid & 31;
  const int l15 = lane & 15, khalf = (lane >> 4) << 1;
  float* sbase = states + ((size_t)b * LSEQ + c * TT) * NS;
  for (int i = 0; i < 2; ++i) {
    const int tile = wave * 2 + i;           // 16 tiles over 8 waves
    const int mTile = tile >> 2, nTile = tile & 3;
    v8f acc = {};
    for (int k0 = 0; k0 < TT; k0 += 4) {
      const int kb = k0 + khalf;
      v2f a, bb;
      const int tr = mTile * 16 + l15;
      a.x = Um[tr][kb];    a.y = Um[tr][kb + 1];
      const int col = nTile * 16 + l15;
      bb.x = Vm[kb][col];  bb.y = Vm[kb + 1][col];
      acc = wmma4(a, bb, acc);
    }
    const int tofs = (lane >= 16) ? 8 : 0;
    for (int r = 0; r < 8; ++r) {
      const int t = mTile * 16 + r + tofs;
      const int n = nTile * 16 + l15;
      sbase[t * NS + n] = acc[r];
    }
  }
}

// ------------------------------------------------- initial-state correction --
// states[b, c*64+t, n] += (A^t h0_c[b])[n]  via  H0_c(32x64) @ PT(64x4096).
__global__ void lssm_init_correct(const float* __restrict__ H0, const float* __restrict__ PT,
                                  float* __restrict__ states) {
  const int c = blockIdx.x + 1;  // chunk 0 has h0 = 0
  __shared__ float h0s[NB][PAD];
  const int tid = threadIdx.x;
  for (int i = tid; i < NB * NS; i += 256)
    h0s[i >> 6][i & 63] = H0[(size_t)c * (NB * NS) + i];
  __syncthreads();
  const int wave = tid >> 5, lane = tid & 31;
  const int l15 = lane & 15, khalf = (lane >> 4) << 1;
  // 512 tiles: mTile in {0,1} over batch, nTile in [0,256) over t*64+n columns
  for (int tile = wave; tile < 512; tile += 8) {
    const int mTile = tile & 1, nTile = tile >> 1;
    const float* ptc = PT + nTile * 16 + l15;
    __builtin_prefetch(ptc + 16 * 4, 0, 0);  // pull upcoming PT columns toward GL2/L0
    v8f acc = {};
    for (int k0 = 0; k0 < NS; k0 += 4) {
      const int kb = k0 + khalf;
      v2f a, bb;
      const int br = mTile * 16 + l15;
      a.x = h0s[br][kb];        a.y = h0s[br][kb + 1];
      bb.x = ptc[kb * (TT * NS)];
      bb.y = ptc[(kb + 1) * (TT * NS)];
      acc = wmma4(a, bb, acc);
    }
    const int bofs = (lane >= 16) ? 8 : 0;
    const int colbase = nTile * 16 + l15;
    for (int r = 0; r < 8; ++r) {
      const int b = mTile * 16 + r + bofs;
      float* p = states + (size_t)b * LSEQ * NS + (size_t)c * (TT * NS) + colbase;
      *p += acc[r];
    }
  }
}

// ----------------------------------------------------------- out projection --
// out[b,l] = states[b,l,:] . C + D*u[b,l]
__global__ void lssm_outproj(const float* __restrict__ u, const float* __restrict__ states,
                             const float* __restrict__ Cw, const float* __restrict__ Dw,
                             float* __restrict__ out) {
  __shared__ float cs[NS];
  const int tid = threadIdx.x;
  if (tid < NS) cs[tid] = Cw[tid];
  __syncthreads();
  const size_t gid = (size_t)blockIdx.x * blockDim.x + tid;  // over B*L
  const float d = Dw[0];
  const float4* row = (const float4*)(states + gid * NS);
  float s = 0.f;
#pragma unroll
  for (int i = 0; i < NS / 4; ++i) {
    float4 v = row[i];
    s += v.x * cs[i * 4 + 0] + v.y * cs[i * 4 + 1] + v.z * cs[i * 4 + 2] + v.w * cs[i * 4 + 3];
  }
  out[gid] = s + d * u[gid];
}

extern "C" void kernel_launch(void* const* d_in, const int* in_sizes, int n_in,
                              void* d_out, int out_size, void* d_ws, size_t ws_size,
                              hipStream_t stream) {
  const float* u  = (const float*)d_in[0];
  const float* A  = (const float*)d_in[1];
  const float* Bw = (const float*)d_in[2];
  const float* Cw = (const float*)d_in[3];
  const float* Dw = (const float*)d_in[4];

  float* out    = (float*)d_out;
  float* states = out + (size_t)NB * LSEQ;       // tuple output #2

  float* ws   = (float*)d_ws;
  float* Vmat = ws;                              // 4096
  float* PT   = Vmat + NS * NS;                  // 262144
  float* A64  = PT + NS * (TT * NS);             // 4096
  float* S    = A64 + NS * NS;                   // 524288
  float* H0   = S + (size_t)NCH * NB * NS;       // 524288

  lssm_setup<<<1, 256, 0, stream>>>(A, Bw, Vmat, PT, A64);
  lssm_chunksums<<<NCH, 256, 0, stream>>>(u, Vmat, S);
  lssm_carry<<<1, 256, 0, stream>>>(A64, S, H0);
  lssm_local<<<dim3(NCH, NB), 256, 0, stream>>>(u, Vmat, states);
  lssm_init_correct<<<NCH - 1, 256, 0, stream>>>(H0, PT, states);
  lssm_outproj<<<(NB * LSEQ) / 256, 256, 0, stream>>>(u, states, Cw, Dw, out);
}